// WeightedSkillSAGE_38955353375249
// MI455X (gfx1250) — compile-verified
//
#include <hip/hip_runtime.h>

// ---------------------------------------------------------------------------
// WeightedSkillSAGE on gfx1250 (MI455X, wave32, WMMA).
// GEMMs use V_WMMA_F32_16X16X4_F32 (full fp32 precision; workload is
// bandwidth-bound on the edge scatter-max, so fp32 matrix ops are free).
// segment_max implemented as global_atomic_max_u32 on float bit patterns
// (valid because all messages are >= +0.0) with zero-initialized output,
// which also implements the "empty segment -> 0" rule.
// ---------------------------------------------------------------------------

typedef __attribute__((ext_vector_type(2))) float v2f;
typedef __attribute__((ext_vector_type(8))) float v8f;

// out[row, co + j] (row stride ow) = act(A[row,:128] @ W[128,KO] + bias[j])
// Requires nrows % 16 == 0 (true here: 50000 = 3125*16) so EXEC is all-ones
// at every WMMA. 8 waves per block; each wave computes one 16x16 tile.
template <int KO>
__global__ __launch_bounds__(256) void gemm_wmma_f32(
    const float* __restrict__ A, const float* __restrict__ W,
    const float* __restrict__ bias, float* __restrict__ out,
    int nrows, int ow, int co, int dorelu) {
  __shared__ float Wl[128 * KO];  // <= 64 KB, well inside 320 KB WGP LDS
  for (int i = threadIdx.x; i < 128 * KO; i += 256) Wl[i] = W[i];
  __syncthreads();

  const int lane = threadIdx.x & 31;
  const int wave = threadIdx.x >> 5;
  const int nct = KO >> 4;                 // column tiles
  const int tile = blockIdx.x * 8 + wave;
  const int row_tile = tile / nct;
  const int col_tile = tile - row_tile * nct;
  if (row_tile * 16 >= nrows) return;      // wave-uniform exit (no barriers after)

  const int half = lane >> 4;              // 0: K={0,1}, 1: K={2,3} (A layout)
  const int lrow = lane & 15;
  const int r = row_tile * 16 + lrow;      // A row / output column index role
  const int colbase = col_tile * 16;

  const float* __restrict__ arow = A + (size_t)r * 128;
  v8f acc = {0.f, 0.f, 0.f, 0.f, 0.f, 0.f, 0.f, 0.f};
#pragma unroll
  for (int kk = 0; kk < 32; ++kk) {
    const int k = kk * 4 + half * 2;
    v2f a;
    a.x = arow[k];
    a.y = arow[k + 1];
    v2f b;
    b.x = Wl[k * KO + colbase + lrow];
    b.y = Wl[(k + 1) * KO + colbase + lrow];
    acc = __builtin_amdgcn_wmma_f32_16x16x4_f32(false, a, false, b, (short)0,
                                                acc, false, false);
  }

  const float bv = bias[colbase + lrow];
#pragma unroll
  for (int v = 0; v < 8; ++v) {
    float x = acc[v] + bv;
    if (dorelu) x = fmaxf(x, 0.0f);
    // D layout: lanes 0-15 -> M = v, lanes 16-31 -> M = v + 8; N = lrow.
    out[(size_t)(row_tile * 16 + v + half * 8) * ow + co + colbase + lrow] = x;
  }
}

__global__ __launch_bounds__(256) void zero_f32(float* __restrict__ p,
                                                long long n) {
  long long i = (long long)blockIdx.x * blockDim.x + threadIdx.x;
  const long long stride = (long long)gridDim.x * blockDim.x;
  for (; i < n; i += stride) p[i] = 0.0f;
}

// One wave per edge: gather 128 floats of m[src], scale by ew, scatter-max
// into agg[dst] via u32 atomics (all values >= +0.0f).
__global__ __launch_bounds__(256) void edge_segmax(
    const float* __restrict__ m, const int* __restrict__ ei,
    const float* __restrict__ ew, unsigned int* __restrict__ agg, int E) {
  const int lane = threadIdx.x & 31;
  const int e = blockIdx.x * 8 + (threadIdx.x >> 5);
  if (e >= E) return;
  const int src = ei[e];
  const int dst = ei[E + e];
  const float w = ew[e];
  const float4 v = *(const float4*)(m + (size_t)src * 128 + lane * 4);
  unsigned int* ap = agg + (size_t)dst * 128 + lane * 4;
  atomicMax(ap + 0, __float_as_uint(v.x * w));
  atomicMax(ap + 1, __float_as_uint(v.y * w));
  atomicMax(ap + 2, __float_as_uint(v.z * w));
  atomicMax(ap + 3, __float_as_uint(v.w * w));
}

// One wave per row: y = relu(t_row); sum_row += y / max(||y||, eps)
__global__ __launch_bounds__(256) void relu_norm_accum(
    const float* __restrict__ t, float* __restrict__ sum, int nrows) {
  const int lane = threadIdx.x & 31;
  const int row = blockIdx.x * 8 + (threadIdx.x >> 5);
  if (row >= nrows) return;
  float4 v = *(const float4*)(t + (size_t)row * 128 + lane * 4);
  v.x = fmaxf(v.x, 0.0f);
  v.y = fmaxf(v.y, 0.0f);
  v.z = fmaxf(v.z, 0.0f);
  v.w = fmaxf(v.w, 0.0f);
  float s = v.x * v.x + v.y * v.y + v.z * v.z + v.w * v.w;
#pragma unroll
  for (int o = 16; o > 0; o >>= 1) s += __shfl_xor(s, o);
  const float inv = 1.0f / fmaxf(sqrtf(s), 1e-12f);
  float* sp = sum + (size_t)row * 128 + lane * 4;
  float4 a = *(const float4*)sp;
  a.x += v.x * inv;
  a.y += v.y * inv;
  a.z += v.z * inv;
  a.w += v.w * inv;
  *(float4*)sp = a;
}

// ---------------------------------------------------------------------------

static void launch_gemm(const float* A, const float* W, const float* b,
                        float* out, int nrows, int KO, int ow, int co,
                        int relu, hipStream_t s) {
  const int tiles = (nrows / 16) * (KO / 16);
  const int blocks = (tiles + 7) / 8;
  if (KO == 128)
    gemm_wmma_f32<128><<<blocks, 256, 0, s>>>(A, W, b, out, nrows, ow, co, relu);
  else
    gemm_wmma_f32<64><<<blocks, 256, 0, s>>>(A, W, b, out, nrows, ow, co, relu);
}

static void launch_zero(float* p, long long n, hipStream_t s) {
  int blocks = (int)((n + 255) / 256);
  if (blocks > 16384) blocks = 16384;
  zero_f32<<<blocks, 256, 0, s>>>(p, n);
}

extern "C" void kernel_launch(void* const* d_in, const int* in_sizes, int n_in,
                              void* d_out, int out_size, void* d_ws,
                              size_t ws_size, hipStream_t stream) {
  (void)n_in; (void)out_size; (void)ws_size;
  const int HID = 128;
  const int Ns = in_sizes[0] / HID;   // 50000
  const int Nj = in_sizes[1] / HID;   // 50000

  const float* x_skill = (const float*)d_in[0];
  const float* x_job = (const float*)d_in[1];

  // params flattened as JAX pytree (dict keys sorted):
  // b_in_job, b_in_skill, b_out_job, b_out_skill,
  // jj{bi,bj,bp,wi,wj,wp}, req{...}, rev_req{...}, ss{...},
  // w_in_job, w_in_skill, w_out_job, w_out_skill
  const int P = 14;
  const float* b_in_job    = (const float*)d_in[P + 0];
  const float* b_in_skill  = (const float*)d_in[P + 1];
  const float* b_out_job   = (const float*)d_in[P + 2];
  const float* b_out_skill = (const float*)d_in[P + 3];
  const int CJJ = P + 4, CREQ = P + 10, CRR = P + 16, CSS = P + 22;
  const float* w_in_job    = (const float*)d_in[P + 28];
  const float* w_in_skill  = (const float*)d_in[P + 29];
  const float* w_out_job   = (const float*)d_in[P + 30];
  const float* w_out_skill = (const float*)d_in[P + 31];

  // workspace layout (floats)
  const size_t szS = (size_t)Ns * HID;
  const size_t szJ = (size_t)Nj * HID;
  const size_t szM = (szS > szJ) ? szS : szJ;
  float* ws = (float*)d_ws;
  float* hs    = ws;                 ws += szS;
  float* hj    = ws;                 ws += szJ;
  float* sum_s = ws;                 ws += szS;
  float* sum_j = ws;                 ws += szJ;
  float* m     = ws;                 ws += szM;
  float* agg   = ws;                 ws += szM;
  float* tmp   = ws;                 ws += szM;

  // 1) input transforms
  launch_gemm(x_skill, w_in_skill, b_in_skill, hs, Ns, 128, 128, 0, 1, stream);
  launch_gemm(x_job,   w_in_job,   b_in_job,   hj, Nj, 128, 128, 0, 1, stream);
  launch_zero(sum_s, (long long)szS, stream);
  launch_zero(sum_j, (long long)szJ, stream);

  // 2) six relations: (ei idx, ew idx, src feats, n_src, dst feats, sum buf,
  //                    n_dst, conv params, reuse m from previous relation?)
  struct Rel {
    int iei, iew;
    const float* hsrc; int nsrc;
    const float* hdst; float* sumdst; int ndst;
    int conv; int reuse_m;
  };
  const Rel rels[6] = {
      {2,  3,  hj, Nj, hs, sum_s, Ns, CREQ, 0},  // req:     job  -> skill
      {4,  5,  hs, Ns, hj, sum_j, Nj, CRR,  0},  // rev_req: skill-> job
      {6,  7,  hs, Ns, hs, sum_s, Ns, CSS,  0},  // ss
      {8,  9,  hs, Ns, hs, sum_s, Ns, CSS,  1},  // rev_ss (shared conv -> reuse m)
      {10, 11, hj, Nj, hj, sum_j, Nj, CJJ,  0},  // jj
      {12, 13, hj, Nj, hj, sum_j, Nj, CJJ,  1},  // rev_jj (shared conv -> reuse m)
  };

  for (int ri = 0; ri < 6; ++ri) {
    const Rel& R = rels[ri];
    const int* ei = (const int*)d_in[R.iei];
    const float* ew = (const float*)d_in[R.iew];
    const int E = in_sizes[R.iew];
    const float* bi = (const float*)d_in[R.conv + 0];
    const float* bj = (const float*)d_in[R.conv + 1];
    const float* bp = (const float*)d_in[R.conv + 2];
    const float* wi = (const float*)d_in[R.conv + 3];
    const float* wj = (const float*)d_in[R.conv + 4];
    const float* wp = (const float*)d_in[R.conv + 5];

    launch_zero(agg, (long long)R.ndst * HID, stream);
    if (!R.reuse_m)  // m = relu(hsrc @ wp + bp)
      launch_gemm(R.hsrc, wp, bp, m, R.nsrc, 128, 128, 0, 1, stream);
    edge_segmax<<<(E + 7) / 8, 256, 0, stream>>>(m, ei, ew,
                                                 (unsigned int*)agg, E);
    // tmp[:,0:64]  = hdst @ wi + bi ; tmp[:,64:128] = agg @ wj + bj
    launch_gemm(R.hdst, wi, bi, tmp, R.ndst, 64, 128, 0, 0, stream);
    launch_gemm(agg,    wj, bj, tmp, R.ndst, 64, 128, 64, 0, stream);
    relu_norm_accum<<<(R.ndst + 7) / 8, 256, 0, stream>>>(tmp, R.sumdst,
                                                          R.ndst);
  }

  // 3) output transforms -> d_out = [out_s | out_j]
  float* out_s = (float*)d_out;
  float* out_j = out_s + (size_t)Ns * HID;
  launch_gemm(sum_s, w_out_skill, b_out_skill, out_s, Ns, 128, 128, 0, 1, stream);
  launch_gemm(sum_j, w_out_job,   b_out_job,   out_j, Nj, 128, 128, 0, 1, stream);
}